// HungarianMatcher_20186346291680
// MI455X (gfx1250) — compile-verified
//
#include <hip/hip_runtime.h>
#include <hip/hip_bf16.h>
#include <math.h>

// Problem constants (fixed by reference setup_inputs)
#define BB 2
#define QQ 900
#define CCLS 91
#define TT 64
#define HH 128
#define WWD 128
#define PP 1024          // (128*128)/16 sample points
#define ALPHA_C 0.25f
#define BKC 32           // K-chunk per LDS stage (per wave)
#define KPW (PP / 4)     // K points per wave (4 waves split K)

typedef __attribute__((ext_vector_type(2))) float v2f;
typedef __attribute__((ext_vector_type(8))) float v8f;

__device__ __forceinline__ float sigmoidf_(float x) { return 1.0f / (1.0f + expf(-x)); }
// softplus(x) = -log(sigmoid(-x)) = log(1+e^x), numerically stable
__device__ __forceinline__ float softplusf_(float x) {
    return (x > 0.0f) ? (x + log1pf(expf(-x))) : log1pf(expf(x));
}
// Order-preserving encoding of float into uint for atomicMax
__device__ __forceinline__ unsigned encodeKey(float x) {
    unsigned u = __float_as_uint(x);
    return (u & 0x80000000u) ? ~u : (u | 0x80000000u);
}
__device__ __forceinline__ float decodeKey(unsigned e) {
    unsigned u = (e & 0x80000000u) ? (e ^ 0x80000000u) : ~e;
    return __uint_as_float(u);
}

// GIoU of one (pred cxcywh, tgt cxcywh) pair
__device__ __forceinline__ float giou_pair(const float* pb, const float* tb) {
    float px0 = pb[0] - 0.5f * pb[2], py0 = pb[1] - 0.5f * pb[3];
    float px1 = pb[0] + 0.5f * pb[2], py1 = pb[1] + 0.5f * pb[3];
    float tx0 = tb[0] - 0.5f * tb[2], ty0 = tb[1] - 0.5f * tb[3];
    float tx1 = tb[0] + 0.5f * tb[2], ty1 = tb[1] + 0.5f * tb[3];
    float area1 = (px1 - px0) * (py1 - py0);
    float area2 = (tx1 - tx0) * (ty1 - ty0);
    float iw = fmaxf(fminf(px1, tx1) - fmaxf(px0, tx0), 0.0f);
    float ih = fmaxf(fminf(py1, ty1) - fmaxf(py0, ty0), 0.0f);
    float inter = iw * ih;
    float uni = area1 + area2 - inter;
    float iou = inter / uni;
    float cw = fmaxf(fmaxf(px1, tx1) - fminf(px0, tx0), 0.0f);
    float ch = fmaxf(fmaxf(py1, ty1) - fminf(py0, ty0), 0.0f);
    float areac = cw * ch;
    return iou - (areac - uni) / areac;
}

__global__ void init_max(unsigned* maxbuf) {
    if (threadIdx.x < BB) maxbuf[threadIdx.x] = 0x007FFFFFu; // encodeKey(-inf)
}

// Nearest-sample tgt masks at point_coords -> tptsT[b][p][t] (B-matrix layout) + per-target sums
__global__ __launch_bounds__(128)
void prep_tgt(const float* __restrict__ tmasks, const float* __restrict__ pcoords,
              float* __restrict__ tptsT, float* __restrict__ tsum) {
    __shared__ float red[128];
    const int t = blockIdx.x, b = blockIdx.y, tid = threadIdx.x;
    const float* img = tmasks + ((size_t)(b * TT + t)) * (HH * WWD);
    float s = 0.0f;
#pragma unroll
    for (int i = 0; i < PP / 128; ++i) {
        int p = tid + i * 128;
        float px = pcoords[2 * p], py = pcoords[2 * p + 1];
        float x = rintf(px * (float)WWD - 0.5f);   // round-half-even == jnp.round
        float y = rintf(py * (float)HH - 0.5f);
        float val = 0.0f;
        if (x >= 0.0f && x < (float)WWD && y >= 0.0f && y < (float)HH)
            val = img[(int)y * WWD + (int)x];
        tptsT[((size_t)b * PP + p) * TT + t] = val;
        s += val;
    }
    red[tid] = s;
    __syncthreads();
    for (int off = 64; off > 0; off >>= 1) {
        if (tid < off) red[tid] += red[tid + off];
        __syncthreads();
    }
    if (tid == 0) tsum[b * TT + t] = red[0];
}

// Main fused kernel: each block owns a 16-row Q tile; the 4 waves split the K
// (point) dimension 4-ways for occupancy, each running its own bilinear-sample
// + two f32 WMMA GEMMs (raw dot & sigmoid dot); partials reduced through LDS,
// wave 0 fuses bbox/class/GIoU costs, writes C, tracks per-batch finite max.
__global__ __launch_bounds__(128)
void matcher_main(const float* __restrict__ logits,
                  const float* __restrict__ pboxes,
                  const float* __restrict__ pmasks,
                  const float* __restrict__ tboxes,
                  const int*   __restrict__ tlabels,
                  const float* __restrict__ pcoords,
                  const float* __restrict__ tptsT,
                  const float* __restrict__ tsum,
                  unsigned*    __restrict__ maxbuf,
                  float*       __restrict__ out) {
    __shared__ float As[4][16][BKC + 4];  // per-wave A tile (pad -> conflict-free)
    __shared__ float Bs[4][BKC][TT];      // per-wave B tile; reused as reduction scratch
    __shared__ float rowNegS[4][16];      // per-wave per-row softplus sums
    __shared__ float rowSigS[4][16];      // per-wave per-row sigmoid sums

    const int b = blockIdx.y;
    const int tid = threadIdx.x;
    const int wave = tid >> 5;
    const int lane = tid & 31;
    const int m = lane & 15;          // row within 16x16 tile
    const int half = lane >> 4;       // lane half selects K pair (0,1) vs (2,3)
    const int qbase = blockIdx.x * 16;
    const int qrow = qbase + m;       // row this lane samples
    const bool rowValid = (qrow < QQ);
    const float* mymask = pmasks + ((size_t)(b * QQ + (rowValid ? qrow : 0))) * (HH * WWD);

    const int kBeg = wave * KPW;      // this wave's 256-point K range
    const int kEnd = kBeg + KPW;

    v8f dotAcc[4] = {};   // pred_pts . tgt_pts   (4 N-tiles of 16)
    v8f diceAcc[4] = {};  // sigmoid(pred_pts) . tgt_pts
    float negPart = 0.0f, sigPart = 0.0f;

    for (int kc = kBeg; kc < kEnd; kc += BKC) {
        __syncthreads();
        // prefetch next chunk's coords + B slab (global_prefetch_b8)
        if (kc + BKC < kEnd) {
            __builtin_prefetch(&pcoords[2 * (kc + BKC)], 0, 1);
            __builtin_prefetch(&tptsT[((size_t)b * PP + kc + BKC) * TT], 0, 1);
        }
        // stage this wave's B chunk [BKC x TT] (coalesced, per-wave region)
#pragma unroll 8
        for (int i = 0; i < (BKC * TT) / 32; ++i) {
            int e = lane + i * 32;
            int pl = e >> 6, t = e & 63;
            Bs[wave][pl][t] = tptsT[((size_t)b * PP + kc + pl) * TT + t];
        }
        // bilinear-sample A chunk: lane covers row m, cols half*16 .. half*16+15
#pragma unroll 4
        for (int j = 0; j < BKC / 2; ++j) {
            int pl = half * (BKC / 2) + j;
            int p = kc + pl;
            float val = 0.0f;
            if (rowValid) {
                float px = pcoords[2 * p], py = pcoords[2 * p + 1];
                float x = px * (float)WWD - 0.5f;
                float y = py * (float)HH - 0.5f;
                float x0 = floorf(x), y0 = floorf(y);
                float wx1 = x - x0, wx0 = 1.0f - wx1;
                float wy1 = y - y0, wy0 = 1.0f - wy1;
                int xi = (int)x0, yi = (int)y0;
                float v00 = 0.f, v01 = 0.f, v10 = 0.f, v11 = 0.f;
                bool xok0 = (xi >= 0) & (xi < WWD);
                bool xok1 = (xi + 1 >= 0) & (xi + 1 < WWD);
                if (yi >= 0 && yi < HH) {
                    if (xok0) v00 = mymask[yi * WWD + xi];
                    if (xok1) v01 = mymask[yi * WWD + xi + 1];
                }
                if (yi + 1 >= 0 && yi + 1 < HH) {
                    if (xok0) v10 = mymask[(yi + 1) * WWD + xi];
                    if (xok1) v11 = mymask[(yi + 1) * WWD + xi + 1];
                }
                val = v00 * wy0 * wx0 + v01 * wy0 * wx1 + v10 * wy1 * wx0 + v11 * wy1 * wx1;
            }
            As[wave][m][pl] = val;
            negPart += softplusf_(val);   // neg_ce = softplus(x); rows >= Q never used
            sigPart += sigmoidf_(val);
        }
        __syncthreads();
        // K loop: V_WMMA_F32_16X16X4_F32, A 16x4 / B 4x16 fragments per ISA layout
#pragma unroll 4
        for (int k0 = 0; k0 < BKC; k0 += 4) {
            const int kk = k0 + (half << 1);  // lanes 0-15: K 0..1, lanes 16-31: K 2..3
            v2f a, asg;
            a[0] = As[wave][m][kk];
            a[1] = As[wave][m][kk + 1];
            asg[0] = sigmoidf_(a[0]);
            asg[1] = sigmoidf_(a[1]);
#pragma unroll
            for (int nt = 0; nt < 4; ++nt) {
                v2f bf;
                bf[0] = Bs[wave][kk][nt * 16 + m];
                bf[1] = Bs[wave][kk + 1][nt * 16 + m];
                dotAcc[nt] = __builtin_amdgcn_wmma_f32_16x16x4_f32(
                    false, a, false, bf, (short)0, dotAcc[nt], false, false);
                diceAcc[nt] = __builtin_amdgcn_wmma_f32_16x16x4_f32(
                    false, asg, false, bf, (short)0, diceAcc[nt], false, false);
            }
        }
    }

    // per-wave row sums: combine lane pair (l, l^16)
    float negRow = negPart + __shfl_xor(negPart, 16, 32);
    float sigRow = sigPart + __shfl_xor(sigPart, 16, 32);

    __syncthreads();                       // all waves done with As/Bs
    if (lane < 16) {
        rowNegS[wave][lane] = negRow;
        rowSigS[wave][lane] = sigRow;
    }
    // waves 1..3 dump accumulators into LDS (reuse Bs; stride 65 -> conflict-free)
    float* red = &Bs[0][0][0];             // 4*32*64 = 8192 floats available
    if (wave != 0) {
        float* dst = red + ((wave - 1) * 32 + lane) * 65;
#pragma unroll
        for (int nt = 0; nt < 4; ++nt)
#pragma unroll
            for (int r = 0; r < 8; ++r) {
                dst[nt * 8 + r] = dotAcc[nt][r];
                dst[32 + nt * 8 + r] = diceAcc[nt][r];
            }
    }
    __syncthreads();

    if (wave == 0) {
#pragma unroll
        for (int w = 1; w < 4; ++w) {
            const float* src = red + ((w - 1) * 32 + lane) * 65;
#pragma unroll
            for (int nt = 0; nt < 4; ++nt)
#pragma unroll
                for (int r = 0; r < 8; ++r) {
                    dotAcc[nt][r] += src[nt * 8 + r];
                    diceAcc[nt][r] += src[32 + nt * 8 + r];
                }
        }

        float localBest = -INFINITY;
        bool any = false;
#pragma unroll
        for (int nt = 0; nt < 4; ++nt) {
#pragma unroll
            for (int r = 0; r < 8; ++r) {
                const int mrow = r + (half << 3);  // C/D layout: VGPR r -> M=r / M=r+8
                const float ns = rowNegS[0][mrow] + rowNegS[1][mrow] +
                                 rowNegS[2][mrow] + rowNegS[3][mrow];
                const float ss = rowSigS[0][mrow] + rowSigS[1][mrow] +
                                 rowSigS[2][mrow] + rowSigS[3][mrow];
                const int q = qbase + mrow;
                const int t = nt * 16 + m;
                const float dotv = dotAcc[nt][r];
                const float dicev = diceAcc[nt][r];
                if (q < QQ) {
                    // mask CE: (sum_p neg_ce - <x, t>) / P
                    float cce = (ns - dotv) * (1.0f / (float)PP);
                    // dice
                    float tsv = tsum[b * TT + t];
                    float cdice = 1.0f - (2.0f * dicev + 1.0f) / (ss + tsv + 1.0f);
                    // bbox L1 + GIoU
                    const float* pb = pboxes + ((size_t)(b * QQ + q)) * 4;
                    const float* tb = tboxes + ((size_t)(b * TT + t)) * 4;
                    float cbbox = fabsf(pb[0] - tb[0]) + fabsf(pb[1] - tb[1]) +
                                  fabsf(pb[2] - tb[2]) + fabsf(pb[3] - tb[3]);
                    float g = giou_pair(pb, tb);
                    // focal class cost at target label
                    int lab = tlabels[b * TT + t];
                    float lx = logits[((size_t)(b * QQ + q)) * CCLS + lab];
                    float pr = sigmoidf_(lx);
                    float ccls = ALPHA_C * (1.0f - pr) * (1.0f - pr) * softplusf_(-lx) -
                                 (1.0f - ALPHA_C) * pr * pr * softplusf_(lx);
                    float c = cbbox + ccls - g + cce + cdice;
                    out[((size_t)b * QQ + q) * TT + t] = c;
                    float contrib = __builtin_isfinite(c) ? c : 0.0f;
                    localBest = fmaxf(localBest, contrib);
                    any = true;
                }
            }
        }
        if (any) atomicMax(&maxbuf[b], encodeKey(localBest));
    }
}

__global__ void fixup(float* __restrict__ out, const unsigned* __restrict__ maxbuf) {
    int idx = blockIdx.x * blockDim.x + threadIdx.x;
    if (idx >= BB * QQ * TT) return;
    float c = out[idx];
    if (!__builtin_isfinite(c)) {
        int b = idx / (QQ * TT);
        out[idx] = 2.0f * decodeKey(maxbuf[b]);
    }
}

extern "C" void kernel_launch(void* const* d_in, const int* in_sizes, int n_in,
                              void* d_out, int out_size, void* d_ws, size_t ws_size,
                              hipStream_t stream) {
    const float* logits  = (const float*)d_in[0];  // [B,Q,C]
    const float* pboxes  = (const float*)d_in[1];  // [B,Q,4]
    const float* pmasks  = (const float*)d_in[2];  // [B,Q,H,W]
    const float* tboxes  = (const float*)d_in[3];  // [B,T,4]
    const float* tmasks  = (const float*)d_in[4];  // [B,T,H,W]
    const float* pcoords = (const float*)d_in[5];  // [1,P,2]
    const int*   tlabels = (const int*)d_in[6];    // [B,T]
    float* out = (float*)d_out;                    // [B,Q,T]

    // workspace: tptsT [B,P,T] (512KB) + tsum [B,T] + maxbuf [B]
    float* tptsT = (float*)d_ws;
    float* tsum = tptsT + (size_t)BB * PP * TT;
    unsigned* maxbuf = (unsigned*)(tsum + BB * TT);

    hipLaunchKernelGGL(init_max, dim3(1), dim3(32), 0, stream, maxbuf);
    hipLaunchKernelGGL(prep_tgt, dim3(TT, BB), dim3(128), 0, stream,
                       tmasks, pcoords, tptsT, tsum);
    hipLaunchKernelGGL(matcher_main, dim3((QQ + 15) / 16, BB), dim3(128), 0, stream,
                       logits, pboxes, pmasks, tboxes, tlabels, pcoords,
                       tptsT, tsum, maxbuf, out);
    const int n = BB * QQ * TT;
    hipLaunchKernelGGL(fixup, dim3((n + 255) / 256), dim3(256), 0, stream, out, maxbuf);
}